// SoftHd_3994319585319
// MI455X (gfx1250) — compile-verified
//
#include <hip/hip_runtime.h>
#include <stdint.h>

typedef __attribute__((ext_vector_type(2))) float v2f;
typedef __attribute__((ext_vector_type(8))) float v8f;

#define NN  256   // nodes per graph
#define DF  32    // feature dim
#define LDA 36    // padded LDS row stride (floats): 16B-aligned, bank-spread

__global__ void softhd_zero_kernel(int* __restrict__ p, int n) {
    int i = blockIdx.x * blockDim.x + threadIdx.x;
    if (i < n) p[i] = 0;
}

__global__ void softhd_degree_kernel(const int* __restrict__ s1, const int* __restrict__ s2,
                                     int* __restrict__ d1, int* __restrict__ d2, int e) {
    int i = blockIdx.x * blockDim.x + threadIdx.x;
    if (i < e) {
        atomicAdd(&d1[s1[i]], 1);
        atomicAdd(&d2[s2[i]], 1);
    }
}

// LDS float atomic-min (no return): ds_min_num_f32. Generic pointer to a
// __shared__ object carries the LDS byte offset in its low 32 bits.
__device__ __forceinline__ void lds_min_f32(float* p, float v) {
    unsigned off = (unsigned)(uintptr_t)p;
    asm volatile("ds_min_num_f32 %0, %1" :: "v"(off), "v"(v) : "memory");
}

__global__ void __launch_bounds__(256)
softhd_main_kernel(const float* __restrict__ x1, const float* __restrict__ x2,
                   const int* __restrict__ deg1, const int* __restrict__ deg2,
                   const int* __restrict__ sz1, const int* __restrict__ sz2,
                   const float* __restrict__ cost, float* __restrict__ out) {
    __shared__ float Xa[NN * LDA];
    __shared__ float Xb[NN * LDA];
    __shared__ float sq1s[NN], sq2s[NN];
    __shared__ float a1[NN], a2[NN];      // c * conn
    __shared__ float colmin[NN], rowm[NN];
    __shared__ float red[8][4];

    const int b    = blockIdx.x;
    const int tid  = threadIdx.x;
    const int lane = tid & 31;
    const int wave = tid >> 5;            // 0..7
    const int hi   = lane >> 4;           // lane half (0/1)
    const int lm   = lane & 15;
    const float c  = cost[0];
    const float INF = 3.4e38f;

    // ---- stage features into LDS (coalesced float4, padded rows) ----
    const float4* g1 = (const float4*)(x1 + (size_t)b * NN * DF);
    const float4* g2 = (const float4*)(x2 + (size_t)b * NN * DF);
    for (int idx = tid; idx < NN * DF / 4; idx += 256) {
        int row = idx >> 3, c4 = idx & 7;
        ((float4*)&Xa[row * LDA])[c4] = g1[idx];
        ((float4*)&Xb[row * LDA])[c4] = g2[idx];
    }
    __syncthreads();

    // ---- squared norms, borders, colmin init ----
    {
        float s1 = 0.f, s2 = 0.f;
        #pragma unroll
        for (int d = 0; d < DF; ++d) {
            float u = Xa[tid * LDA + d]; s1 += u * u;
            float w = Xb[tid * LDA + d]; s2 += w * w;
        }
        sq1s[tid] = s1;
        sq2s[tid] = s2;
        a1[tid] = c * (float)(deg1[b * NN + tid] + 1);
        float e2 = c * (float)(deg2[b * NN + tid] + 1);
        a2[tid] = e2;
        colmin[tid] = e2;                 // appended row pre-seeds column mins
    }
    __syncthreads();

    // ---- fused GEMM (f32 WMMA) + bilateral min tracking ----
    for (int tr = 0; tr < 2; ++tr) {
        const int ti = 2 * wave + tr;     // wave-exclusive tile rows

        // preload A fragments for all 8 k-steps (16x4 f32: M=lm, K=4kk+2hi+r)
        v2f af[8];
        #pragma unroll
        for (int kk = 0; kk < 8; ++kk) {
            int base = (16 * ti + lm) * LDA + kk * 4 + 2 * hi;
            af[kk].x = Xa[base];
            af[kk].y = Xa[base + 1];
        }
        float sq1v[8], rmin[8];
        #pragma unroll
        for (int r = 0; r < 8; ++r) {
            sq1v[r] = sq1s[16 * ti + 8 * hi + r];
            rmin[r] = INF;
        }

        for (int tj = 0; tj < 16; ++tj) {
            v8f acc = {0.f,0.f,0.f,0.f,0.f,0.f,0.f,0.f};
            #pragma unroll
            for (int kk = 0; kk < 8; ++kk) {
                v2f bf;                   // B = X2^T tile: N=lm, K=4kk+2hi+r
                int base = (16 * tj + lm) * LDA + kk * 4 + 2 * hi;
                bf.x = Xb[base];
                bf.y = Xb[base + 1];
                acc = __builtin_amdgcn_wmma_f32_16x16x4_f32(
                        false, af[kk], false, bf, (short)0, acc, false, false);
            }
            // C layout: VGPR r -> (M = r + 8*hi, N = lm)
            float sq2j = sq2s[16 * tj + lm];
            float cl = INF;
            #pragma unroll
            for (int r = 0; r < 8; ++r) {
                float v = sq1v[r] + sq2j - 2.0f * acc[r];
                rmin[r] = fminf(rmin[r], v);
                cl = fminf(cl, v);
            }
            cl = fminf(cl, __shfl_xor(cl, 16, 32));      // combine M halves
            if (hi == 0) lds_min_f32(&colmin[16 * tj + lm], cl);
        }

        // row mins: reduce across the 16 lanes of each half
        #pragma unroll
        for (int r = 0; r < 8; ++r) {
            float v = rmin[r];
            v = fminf(v, __shfl_xor(v, 1, 32));
            v = fminf(v, __shfl_xor(v, 2, 32));
            v = fminf(v, __shfl_xor(v, 4, 32));
            v = fminf(v, __shfl_xor(v, 8, 32));
            rmin[r] = v;
        }
        if (lm == 0) {
            #pragma unroll
            for (int r = 0; r < 8; ++r) {
                int gi = 16 * ti + 8 * hi + r;
                rowm[gi] = fminf(rmin[r], a1[gi]);       // fold border column
            }
        }
    }
    asm volatile("s_wait_dscnt 0" ::: "memory");         // drain asm DS atomics
    __syncthreads();

    // ---- final block reduction ----
    float sr = rowm[tid], sc = colmin[tid], m1 = a1[tid], m2 = a2[tid];
    #pragma unroll
    for (int off = 16; off >= 1; off >>= 1) {
        sr += __shfl_xor(sr, off, 32);
        sc += __shfl_xor(sc, off, 32);
        m1 = fminf(m1, __shfl_xor(m1, off, 32));
        m2 = fminf(m2, __shfl_xor(m2, off, 32));
    }
    if (lane == 0) { red[wave][0] = sr; red[wave][1] = sc;
                     red[wave][2] = m1; red[wave][3] = m2; }
    __syncthreads();
    if (tid == 0) {
        float Sr = 0.f, Sc = 0.f, M1 = INF, M2 = INF;
        #pragma unroll
        for (int w = 0; w < 8; ++w) {
            Sr += red[w][0]; Sc += red[w][1];
            M1 = fminf(M1, red[w][2]); M2 = fminf(M2, red[w][3]);
        }
        float a_sum = Sc + fminf(0.0f, M1);   // last-column min (zero corner)
        float b_sum = Sr + fminf(0.0f, M2);   // last-row min (zero corner)
        out[b] = (a_sum + b_sum) / (float)(sz1[b] + sz2[b]);
    }
}

extern "C" void kernel_launch(void* const* d_in, const int* in_sizes, int n_in,
                              void* d_out, int out_size, void* d_ws, size_t ws_size,
                              hipStream_t stream) {
    const float* x1   = (const float*)d_in[0];
    const float* x2   = (const float*)d_in[1];
    const int*   src1 = (const int*)d_in[2];
    const int*   src2 = (const int*)d_in[3];
    const int*   sz1  = (const int*)d_in[4];
    const int*   sz2  = (const int*)d_in[5];
    const float* cost = (const float*)d_in[6];

    const int N  = in_sizes[0] / DF;   // 131072 total nodes
    const int E  = in_sizes[2];        // 1048576 edges
    const int Bg = in_sizes[4];        // 512 graphs

    int* deg1 = (int*)d_ws;
    int* deg2 = deg1 + N;

    softhd_zero_kernel<<<(2 * N + 255) / 256, 256, 0, stream>>>(deg1, 2 * N);
    softhd_degree_kernel<<<(E + 255) / 256, 256, 0, stream>>>(src1, src2, deg1, deg2, E);
    softhd_main_kernel<<<Bg, 256, 0, stream>>>(x1, x2, deg1, deg2, sz1, sz2, cost,
                                               (float*)d_out);
}